// Net_55946243997874
// MI455X (gfx1250) — compile-verified
//
#include <hip/hip_runtime.h>
#include <hip/hip_bf16.h>
#include <math.h>

// GCN forward on MI455X (gfx1250, wave32).
// norm = D^-1/2 (A+I) D^-1/2 ; h = relu(norm@(xW1)+b1) ; out = log_softmax(norm@(hW2)+b2)
// Aggregations use V_WMMA_F32_16X16X4_F32 with adj as the A (16x4) operand.

#define NK 12288      // nodes
#define FIN 512
#define HID 16
#define NCLS 40
#define NPAD 48       // 40 padded to 3x16 column tiles

typedef __attribute__((ext_vector_type(2))) float v2f;
typedef __attribute__((ext_vector_type(8))) float v8f;

// ---------------- pass 1: degree / d^{-1/2} ----------------
__global__ void deg_kernel(const float* __restrict__ adj, float* __restrict__ d_is) {
    __shared__ float sd[256];
    const int i = blockIdx.x;
    const float* row = adj + (size_t)i * NK;
    float s = 0.f;
    for (int j = threadIdx.x * 4; j < NK; j += 256 * 4) {
        float4 v = *(const float4*)(row + j);
        s += v.x + v.y + v.z + v.w;
    }
    sd[threadIdx.x] = s;
    __syncthreads();
    for (int off = 128; off > 0; off >>= 1) {
        if (threadIdx.x < off) sd[threadIdx.x] += sd[threadIdx.x + off];
        __syncthreads();
    }
    if (threadIdx.x == 0) d_is[i] = rsqrtf(sd[0] + 1.0f);  // +1 self-loop
}

// ---------------- pass 2: xw1sT[c][j] = d_is[j]*(x@W1)[j][c] ----------------
// one wave per 16-row tile; WMMA f32 16x16x4, K = 512
__global__ void xw1_kernel(const float* __restrict__ x, const float* __restrict__ W1,
                           const float* __restrict__ d_is, float* __restrict__ xw1sT) {
    const int m0 = blockIdx.x * 16;
    const int lane = threadIdx.x & 31;
    const int lm = lane & 15, half = lane >> 4;
    const float* aRow = x + (size_t)(m0 + lm) * FIN + 2 * half;
    v8f acc = {};
#pragma unroll 4
    for (int k = 0; k < FIN; k += 4) {
        v2f a = *(const v2f*)(aRow + k);
        const int kb = k + 2 * half;
        v2f b;
        b.x = W1[kb * HID + lm];
        b.y = W1[(kb + 1) * HID + lm];
        acc = __builtin_amdgcn_wmma_f32_16x16x4_f32(false, a, false, b, (short)0, acc,
                                                    false, false);
    }
#pragma unroll
    for (int r = 0; r < 8; ++r) {
        const int m = m0 + r + 8 * half;
        xw1sT[(size_t)lm * NK + m] = acc[r] * d_is[m];
    }
}

// ---------------- shared WMMA aggregation loop ----------------
// acc[nt] += adj[m0:m0+16, kBase:kBase+kLen] @ BT[nt*16:+16][K]^T
template <int NT>
__device__ inline void agg_wmma_loop(const float* __restrict__ adj,
                                     const float* __restrict__ BT, int m0, int kBase,
                                     int kLen, v8f acc[NT]) {
    const int lane = threadIdx.x & 31;
    const int lm = lane & 15, half = lane >> 4;
    const float* aRow = adj + (size_t)(m0 + lm) * NK + 2 * half;
    const float* bRow[NT];
#pragma unroll
    for (int nt = 0; nt < NT; ++nt)
        bRow[nt] = BT + (size_t)(nt * 16 + lm) * NK + 2 * half;
#pragma unroll 4
    for (int k = kBase; k < kBase + kLen; k += 4) {
        v2f a = *(const v2f*)(aRow + k);
#pragma unroll
        for (int nt = 0; nt < NT; ++nt) {
            v2f b = *(const v2f*)(bRow[nt] + k);
            acc[nt] = __builtin_amdgcn_wmma_f32_16x16x4_f32(false, a, false, b, (short)0,
                                                            acc[nt], false, false);
        }
    }
}

// ---------------- pass 3: layer-1 aggregation fused with h@W2 ----------------
// 256 threads = 8 waves; wave w covers K chunk of 1536. Deterministic LDS reduce.
__global__ void agg1_kernel(const float* __restrict__ adj, const float* __restrict__ xw1sT,
                            const float* __restrict__ d_is, const float* __restrict__ b1,
                            const float* __restrict__ W2, float* __restrict__ hw2sT) {
    __shared__ float red[8][16][16];
    __shared__ float hbuf[16][16];
    const int m0 = blockIdx.x * 16;
    const int wave = threadIdx.x >> 5;
    v8f acc = {};
    agg_wmma_loop<1>(adj, xw1sT, m0, wave * (NK / 8), NK / 8, &acc);
    const int lane = threadIdx.x & 31;
    const int lm = lane & 15, half = lane >> 4;
#pragma unroll
    for (int r = 0; r < 8; ++r) red[wave][r + 8 * half][lm] = acc[r];
    __syncthreads();

    const int t = threadIdx.x;
    {   // reduce 8 waves, finish layer 1: h = relu(d_i*(acc + B[i,:]) + b1)
        const int il = t >> 4, c = t & 15;
        float s = 0.f;
#pragma unroll
        for (int w = 0; w < 8; ++w) s += red[w][il][c];
        const int i = m0 + il;
        const float dv = d_is[i];
        float h = dv * (s + xw1sT[(size_t)c * NK + i]) + b1[c];
        hbuf[il][c] = h > 0.f ? h : 0.f;
    }
    __syncthreads();
    // hw2sT[c][i] = d_i * (h[i,:] @ W2)[c], zero-pad c in [40,48)
#pragma unroll
    for (int e = 0; e < 3; ++e) {
        const int idx = t + e * 256;
        const int il = idx / NPAD, c = idx % NPAD;
        const int i = m0 + il;
        float v = 0.f;
        if (c < NCLS) {
            float s = 0.f;
#pragma unroll
            for (int hc = 0; hc < HID; ++hc) s += hbuf[il][hc] * W2[hc * NCLS + c];
            v = d_is[i] * s;
        }
        hw2sT[(size_t)c * NK + i] = v;
    }
}

// ---------------- pass 4: layer-2 aggregation fused with log_softmax ----------------
__global__ void agg2_kernel(const float* __restrict__ adj, const float* __restrict__ hw2sT,
                            const float* __restrict__ d_is, const float* __restrict__ b2,
                            float* __restrict__ out) {
    __shared__ float red[8][16][NPAD];
    __shared__ float lg[16][NCLS];
    const int m0 = blockIdx.x * 16;
    const int wave = threadIdx.x >> 5;
    v8f acc[3] = {};
    agg_wmma_loop<3>(adj, hw2sT, m0, wave * (NK / 8), NK / 8, acc);
    const int lane = threadIdx.x & 31;
    const int lm = lane & 15, half = lane >> 4;
#pragma unroll
    for (int nt = 0; nt < 3; ++nt)
#pragma unroll
        for (int r = 0; r < 8; ++r) red[wave][r + 8 * half][nt * 16 + lm] = acc[nt][r];
    __syncthreads();

    const int t = threadIdx.x;
#pragma unroll
    for (int e = 0; e < 3; ++e) {
        const int idx = t + e * 256;
        const int il = idx / NPAD, c = idx % NPAD;
        if (c < NCLS) {
            float s = 0.f;
#pragma unroll
            for (int w = 0; w < 8; ++w) s += red[w][il][c];
            const int i = m0 + il;
            lg[il][c] = d_is[i] * (s + hw2sT[(size_t)c * NK + i]) + b2[c];
        }
    }
    __syncthreads();
    if (t < 16) {  // per-row log_softmax over 40 classes
        const int i = m0 + t;
        float mx = -INFINITY;
        for (int c = 0; c < NCLS; ++c) mx = fmaxf(mx, lg[t][c]);
        float se = 0.f;
        for (int c = 0; c < NCLS; ++c) se += expf(lg[t][c] - mx);
        const float ls = logf(se) + mx;
        for (int c = 0; c < NCLS; ++c) out[(size_t)i * NCLS + c] = lg[t][c] - ls;
    }
}

extern "C" void kernel_launch(void* const* d_in, const int* in_sizes, int n_in,
                              void* d_out, int out_size, void* d_ws, size_t ws_size,
                              hipStream_t stream) {
    (void)in_sizes; (void)n_in; (void)out_size; (void)ws_size;
    const float* x   = (const float*)d_in[0];   // [12288,512]
    const float* adj = (const float*)d_in[1];   // [12288,12288]
    const float* W1  = (const float*)d_in[2];   // [512,16]
    const float* b1  = (const float*)d_in[3];   // [16]
    const float* W2  = (const float*)d_in[4];   // [16,40]
    const float* b2  = (const float*)d_in[5];   // [40]
    float* out = (float*)d_out;                 // [12288,40]

    float* d_is  = (float*)d_ws;                // 12288
    float* xw1sT = d_is + NK;                   // 16*12288
    float* hw2sT = xw1sT + (size_t)HID * NK;    // 48*12288  (~3.1 MB total)

    deg_kernel<<<NK, 256, 0, stream>>>(adj, d_is);
    xw1_kernel<<<NK / 16, 32, 0, stream>>>(x, W1, d_is, xw1sT);
    agg1_kernel<<<NK / 16, 256, 0, stream>>>(adj, xw1sT, d_is, b1, W2, hw2sT);
    agg2_kernel<<<NK / 16, 256, 0, stream>>>(adj, hw2sT, d_is, b2, out);
}